// TensorLogicSSM_25443386262309
// MI455X (gfx1250) — compile-verified
//
#include <hip/hip_runtime.h>
#include <hip/hip_bf16.h>

#define VOCAB  50257
#define HIDDEN 16
#define BATCH  256
#define SEQLEN 4096
#define VTILES ((VOCAB + 15) / 16)      /* 3142 vocab tiles of 16 rows */
#define NPT    4                        /* vocab tiles per wave in projection */
#define NGROUP ((VTILES + NPT - 1) / NPT) /* 786 */

typedef __attribute__((ext_vector_type(2))) float v2f;
typedef __attribute__((ext_vector_type(8))) float v8f;

__device__ __forceinline__ v2f load2(const float* p) { return *(const v2f*)p; }

// ---------------------------------------------------------------------------
// Kernel 1: embB[v][o] = sum_h embed[v][h] * B_w[o][h] + B_b[o]
// One wave32 per 16-row vocab tile; 4x V_WMMA_F32_16X16X4_F32 over K=16.
// ---------------------------------------------------------------------------
__global__ void embB_wmma_kernel(const float* __restrict__ embed,
                                 const float* __restrict__ B_w,
                                 const float* __restrict__ B_b,
                                 float* __restrict__ embB) {
    const int w = blockIdx.x * (blockDim.x >> 5) + (threadIdx.x >> 5);
    if (w >= VTILES) return;                 // wave-uniform: EXEC stays all-ones
    const int l    = threadIdx.x & 31;
    const int half = l >> 4;                 // K split (A/B) and M split (D)
    const int ln   = l & 15;
    const int vbase = w * 16;
    int arow = vbase + ln; if (arow > VOCAB - 1) arow = VOCAB - 1;  // tail clamp

    const float bias = B_b[ln];              // C(M,N) = B_b[N]
    v8f c;
    #pragma unroll
    for (int r = 0; r < 8; ++r) c[r] = bias;

    #pragma unroll
    for (int kc = 0; kc < 4; ++kc) {
        const int k0 = kc * 4 + 2 * half;
        v2f a = load2(embed + (size_t)arow * HIDDEN + k0);   // A rows = vocab
        v2f b = load2(B_w   + (size_t)ln   * HIDDEN + k0);   // B[k][n] = B_w[n][k]
        c = __builtin_amdgcn_wmma_f32_16x16x4_f32(false, a, false, b,
                                                  (short)0, c, false, false);
    }
    // D(M,N): lane l, reg r -> M = r + 8*half (vocab row), N = ln (hidden)
    if (w < VTILES - 1) {                    // full tile: uniform fast path
        #pragma unroll
        for (int r = 0; r < 8; ++r)
            embB[(size_t)(vbase + r + 8 * half) * HIDDEN + ln] = c[r];
    } else {                                 // last tile: predicated
        #pragma unroll
        for (int r = 0; r < 8; ++r) {
            const int row = vbase + r + 8 * half;
            if (row < VOCAB) embB[(size_t)row * HIDDEN + ln] = c[r];
        }
    }
}

// ---------------------------------------------------------------------------
// Kernel 2: sequential scan  h = sigmoid(Rule*h + bx_t), 4096 steps.
// One batch row per 16 lanes. 4-step chunks; embB gathers lead by 1 chunk,
// seq index loads lead by 2 chunks -> gather latency off the serial path.
// ---------------------------------------------------------------------------
template <bool PRE>
__global__ void scan_kernel(const int*   __restrict__ seq,
                            const float* __restrict__ embB,
                            const float* __restrict__ embed,
                            const float* __restrict__ B_w,
                            const float* __restrict__ B_b,
                            const float* __restrict__ Rule,
                            float* __restrict__ hout) {
    const int ln  = threadIdx.x & 15;
    const int row = blockIdx.x * (blockDim.x >> 4) + (threadIdx.x >> 4);

    float R[16];
    #pragma unroll
    for (int j = 0; j < 16; ++j) R[j] = Rule[ln * 16 + j];

    float Bw[16];
    float bb = 0.0f;
    if (!PRE) {
        #pragma unroll
        for (int j = 0; j < 16; ++j) Bw[j] = B_w[ln * 16 + j];
        bb = B_b[ln];
    }

    const int* sp = seq + (size_t)row * SEQLEN;

    auto bxload = [&](int idx) -> float {
        if (PRE) return embB[(size_t)idx * HIDDEN + ln];
        float s = bb;
        #pragma unroll
        for (int j = 0; j < 16; ++j)
            s = fmaf(Bw[j], embed[(size_t)idx * HIDDEN + j], s);
        return s;
    };

    float h = 0.0f;
    int   idxN[4];                    // indices for chunk c+1
    float bxC[4];                     // bx values for current chunk
    #pragma unroll
    for (int i = 0; i < 4; ++i) idxN[i] = sp[4 + i];
    #pragma unroll
    for (int i = 0; i < 4; ++i) bxC[i] = bxload(sp[i]);

    for (int t0 = 0; t0 < SEQLEN; t0 += 4) {
        // stage 1: gather bx for chunk c+1 (latency hidden behind 4 steps)
        float bxN[4];
        #pragma unroll
        for (int i = 0; i < 4; ++i) bxN[i] = bxload(idxN[i]);
        // stage 2: load indices for chunk c+2 (clamped; tail reads harmless)
        const int b2 = (t0 + 8 <= SEQLEN - 4) ? (t0 + 8) : (SEQLEN - 4);
        #pragma unroll
        for (int i = 0; i < 4; ++i) idxN[i] = sp[b2 + i];
        // stage 3: 4 serial recurrence steps
        #pragma unroll
        for (int i = 0; i < 4; ++i) {
            float a0 = bxC[i], a1 = 0.0f, a2 = 0.0f, a3 = 0.0f;
            #pragma unroll
            for (int j = 0; j < 4; ++j) {
                a0 = fmaf(R[j],      __shfl(h, j,      16), a0);
                a1 = fmaf(R[j + 4],  __shfl(h, j + 4,  16), a1);
                a2 = fmaf(R[j + 8],  __shfl(h, j + 8,  16), a2);
                a3 = fmaf(R[j + 12], __shfl(h, j + 12, 16), a3);
            }
            const float acc = (a0 + a1) + (a2 + a3);
            // sigmoid: v_exp_f32 + v_rcp_f32 (short serial chain)
            h = __builtin_amdgcn_rcpf(1.0f + __expf(-acc));
        }
        #pragma unroll
        for (int i = 0; i < 4; ++i) bxC[i] = bxN[i];
    }

    hout[(size_t)row * HIDDEN + ln] = h;
}

// ---------------------------------------------------------------------------
// Kernel 3: out[b][v] = sum_h H[b][h]*out_w[v][h] + out_b[v]
// One wave32 per 16(batch) x 64(vocab) strip: A loaded once, 4 vocab tiles,
// 16x WMMA_F32_16X16X4_F32, bias folded into C.
// ---------------------------------------------------------------------------
__global__ void proj_wmma_kernel(const float* __restrict__ H,
                                 const float* __restrict__ out_w,
                                 const float* __restrict__ out_b,
                                 float* __restrict__ out) {
    const int w = blockIdx.x * (blockDim.x >> 5) + (threadIdx.x >> 5);
    const int mtile = w & 15;                // 256/16 batch tiles
    const int ng    = w >> 4;                // group of NPT vocab tiles
    if (ng >= NGROUP) return;                // wave-uniform
    const int l    = threadIdx.x & 31;
    const int half = l >> 4;
    const int ln   = l & 15;

    // A tile (H rows for this batch tile), loaded once
    const int hrow = mtile * 16 + ln;
    v2f a[4];
    #pragma unroll
    for (int kc = 0; kc < 4; ++kc)
        a[kc] = load2(H + (size_t)hrow * HIDDEN + kc * 4 + 2 * half);

    #pragma unroll
    for (int s = 0; s < NPT; ++s) {
        const int nt = ng * NPT + s;
        if (nt >= VTILES) break;             // wave-uniform
        const int vb = nt * 16;
        int vcl = vb + ln; if (vcl > VOCAB - 1) vcl = VOCAB - 1;

        const float bias = out_b[vcl];       // C(M,N) = out_b[N]
        v8f c;
        #pragma unroll
        for (int r = 0; r < 8; ++r) c[r] = bias;

        #pragma unroll
        for (int kc = 0; kc < 4; ++kc) {
            v2f b = load2(out_w + (size_t)vcl * HIDDEN + kc * 4 + 2 * half);
            c = __builtin_amdgcn_wmma_f32_16x16x4_f32(false, a[kc], false, b,
                                                      (short)0, c, false, false);
        }
        // D(M,N): M = batch row r+8*half, N = vocab column ln
        if (vb + 16 <= VOCAB) {              // full tile: uniform fast path
            #pragma unroll
            for (int r = 0; r < 8; ++r)
                out[(size_t)(mtile * 16 + r + 8 * half) * VOCAB + vb + ln] = c[r];
        } else {                             // last tile: predicated
            #pragma unroll
            for (int r = 0; r < 8; ++r) {
                const int brow = mtile * 16 + r + 8 * half;
                if (vb + ln < VOCAB) out[(size_t)brow * VOCAB + vb + ln] = c[r];
            }
        }
    }
}

// ---------------------------------------------------------------------------
extern "C" void kernel_launch(void* const* d_in, const int* in_sizes, int n_in,
                              void* d_out, int out_size, void* d_ws, size_t ws_size,
                              hipStream_t stream) {
    const int*   seq   = (const int*)  d_in[0];
    const float* embed = (const float*)d_in[1];
    const float* Rule  = (const float*)d_in[2];
    const float* B_w   = (const float*)d_in[3];
    const float* B_b   = (const float*)d_in[4];
    const float* out_w = (const float*)d_in[5];
    const float* out_b = (const float*)d_in[6];
    float* out  = (float*)d_out;

    float* hbuf = (float*)d_ws;                                   // 256*16 f32
    float* embB = (float*)d_ws + (size_t)BATCH * HIDDEN;          // 50272*16 f32
    const size_t need = ((size_t)BATCH * HIDDEN + (size_t)VTILES * 16 * HIDDEN) * sizeof(float);
    const bool pre = ws_size >= need;

    if (pre) {
        embB_wmma_kernel<<<(VTILES + 7) / 8, 256, 0, stream>>>(embed, B_w, B_b, embB);
        scan_kernel<true><<<BATCH / 16, 256, 0, stream>>>(seq, embB, embed, B_w, B_b, Rule, hbuf);
    } else {
        scan_kernel<false><<<BATCH / 16, 256, 0, stream>>>(seq, nullptr, embed, B_w, B_b, Rule, hbuf);
    }

    // 16 batch tiles * 786 groups = 12576 waves, 8 waves per block -> 1572 blocks
    proj_wmma_kernel<<<(16 * NGROUP) / 8, 256, 0, stream>>>(hbuf, out_w, out_b, out);
}